// ScatteringClassifier_15367392985567
// MI455X (gfx1250) — compile-verified
//
#include <hip/hip_runtime.h>
#include <hip/hip_bf16.h>
#include <math.h>

// Problem constants (match reference)
#define BN 1024   // batch
#define PN 784    // pixels
#define TN 200    // time samples
#define TP 208    // padded time (13 * 16) for WMMA N tiles
#define CN 10     // chunks / classes

typedef float v2f __attribute__((ext_vector_type(2)));
typedef float v8f __attribute__((ext_vector_type(8)));

#define PI2 6.28318530717958647692f

// ---------------------------------------------------------------------------
// Kernel 1: WF = FFT(wf)   (single 200-point DFT; exact angle reduction)
// WF[t] = sum_s (wr + i wi) * e^{-2pi i s t / T}
// ---------------------------------------------------------------------------
__global__ void wf_fft_kernel(const float* __restrict__ wf_re,
                              const float* __restrict__ wf_im,
                              float* __restrict__ WF_re,
                              float* __restrict__ WF_im) {
  int t = threadIdx.x;
  if (t >= TN) return;
  float ar = 0.f, ai = 0.f;
  for (int s = 0; s < TN; ++s) {
    int m = (s * t) % TN;                       // exact periodic reduction
    float th = PI2 * (float)m / (float)TN;
    float cs, sn;
    sincosf(th, &sn, &cs);
    float wr = wf_re[s], wi = wf_im[s];
    // (wr + i wi)(cos - i sin)
    ar += wr * cs + wi * sn;
    ai += wi * cs - wr * sn;
  }
  WF_re[t] = ar;
  WF_im[t] = ai;
}

// ---------------------------------------------------------------------------
// Kernel 2: G[t,u] = (1/T) * WF[t] * e^{+2pi i t u / T}
// ---------------------------------------------------------------------------
__global__ void build_G_kernel(const float* __restrict__ WF_re,
                               const float* __restrict__ WF_im,
                               float* __restrict__ G_re,
                               float* __restrict__ G_im) {
  int idx = blockIdx.x * blockDim.x + threadIdx.x;
  if (idx >= TN * TN) return;
  int t = idx / TN;
  int u = idx % TN;
  int m = (t * u) % TN;
  float th = PI2 * (float)m / (float)TN;
  float cs, sn;
  sincosf(th, &sn, &cs);
  float wr = WF_re[t], wi = WF_im[t];
  const float inv = 1.0f / (float)TN;
  G_re[idx] = (wr * cs - wi * sn) * inv;
  G_im[idx] = (wr * sn + wi * cs) * inv;
}

// ---------------------------------------------------------------------------
// Kernel 3: K[p,u] = sum_t A_re[p,t]*G_re[t,u] - A_im[p,t]*G_im[t,u]
//   A_re = c*W_re - s*W_im ; A_im = c*W_im + s*W_re ; c = v cos(phi), s = v sin(phi)
// One block per pixel p; A row staged in LDS. K padded to TP cols with zeros.
// ---------------------------------------------------------------------------
__global__ void build_K_kernel(const float* __restrict__ v,
                               const float* __restrict__ phase,
                               const float* __restrict__ W_re,
                               const float* __restrict__ W_im,
                               const float* __restrict__ G_re,
                               const float* __restrict__ G_im,
                               float* __restrict__ Kmat) {
  __shared__ float a_re[TN];
  __shared__ float a_im[TN];
  int p = blockIdx.x;
  float cs, sn;
  sincosf(phase[p], &sn, &cs);
  float vc = v[p] * cs;
  float vs = v[p] * sn;
  for (int t = threadIdx.x; t < TN; t += blockDim.x) {
    float wr = W_re[p * TN + t];
    float wi = W_im[p * TN + t];
    a_re[t] = vc * wr - vs * wi;
    a_im[t] = vc * wi + vs * wr;
  }
  __syncthreads();
  for (int u = threadIdx.x; u < TP; u += blockDim.x) {
    float acc = 0.f;
    if (u < TN) {
      for (int t = 0; t < TN; ++t) {
        acc += a_re[t] * G_re[t * TN + u] - a_im[t] * G_im[t * TN + u];
      }
    }
    Kmat[p * TP + u] = acc;   // zero padding for u in [200, 208)
  }
}

// ---------------------------------------------------------------------------
// Kernel 4: real[B,TN] = X[B,PN] @ K[PN,TP]  via V_WMMA_F32_16X16X4_F32
// One wave (32 lanes) per 16x16 output tile. Full fp32 precision.
//
// Fragment layouts (CDNA5 ISA 7.12.2, 32-bit data, wave32):
//   A 16x4 : lanes 0-15 -> M=lane (K=0,1 in v0,v1); lanes 16-31 -> K=2,3
//   B 4x16 : lanes 0-15 -> N=lane (K=0,1 in v0,v1); lanes 16-31 -> K=2,3
//   C/D    : vgpr j, lanes 0-15 -> (M=j, N=lane); lanes 16-31 -> (M=j+8, N=lane-16)
// ---------------------------------------------------------------------------
__global__ void gemm_wmma_kernel(const float* __restrict__ X,
                                 const float* __restrict__ Kmat,
                                 float* __restrict__ realOut) {
  const int lane = threadIdx.x & 31;
  const int r    = lane & 15;
  const int half = lane >> 4;          // 0: K=0,1  1: K=2,3
  const int mBase = blockIdx.x * 16;   // batch tile
  const int nBase = blockIdx.y * 16;   // time tile (padded to 208)

  const float* __restrict__ xrow = X + (size_t)(mBase + r) * PN;
  const float* __restrict__ kcol = Kmat + (nBase + r);

  v8f acc = {};

#pragma unroll 4
  for (int k = 0; k < PN; k += 4) {
    const int kk = k + 2 * half;
    // A fragment: X[mBase+r, kk], X[mBase+r, kk+1]  (contiguous -> 64-bit load)
    v2f a = *(const v2f*)(xrow + kk);
    // B fragment: K[kk, nBase+r], K[kk+1, nBase+r]
    v2f b;
    b.x = kcol[(size_t)kk * TP];
    b.y = kcol[(size_t)(kk + 1) * TP];
    acc = __builtin_amdgcn_wmma_f32_16x16x4_f32(
        /*neg_a=*/false, a, /*neg_b=*/false, b,
        /*c_mod=*/(short)0, acc, /*reuse_a=*/false, /*reuse_b=*/false);
  }

  const int col = nBase + r;
  if (col < TN) {
#pragma unroll
    for (int j = 0; j < 8; ++j) {
      int row = mBase + j + 8 * half;
      realOut[(size_t)row * TN + col] = acc[j];
    }
  }
}

// ---------------------------------------------------------------------------
// Kernel 5: per-batch chunk abs-max + Linear(10,10)
// feats[b,c] = max_{j<20} sqrt(real[b, 20c+j]^2 + 1e-20)
// out[b,c]   = sum_k feats[b,k] * lin_w[c,k] + lin_b[c]
// ---------------------------------------------------------------------------
__global__ void feats_linear_kernel(const float* __restrict__ realB,
                                    const float* __restrict__ lin_w,
                                    const float* __restrict__ lin_b,
                                    float* __restrict__ out) {
  int b = blockIdx.x * blockDim.x + threadIdx.x;
  if (b >= BN) return;
  float f[CN];
#pragma unroll
  for (int c = 0; c < CN; ++c) {
    float m = 0.f;
    for (int j = 0; j < TN / CN; ++j) {
      float x = realB[(size_t)b * TN + c * (TN / CN) + j];
      m = fmaxf(m, sqrtf(x * x + 1e-20f));
    }
    f[c] = m;
  }
#pragma unroll
  for (int c = 0; c < CN; ++c) {
    float acc = lin_b[c];
#pragma unroll
    for (int k = 0; k < CN; ++k) acc += f[k] * lin_w[c * CN + k];
    out[(size_t)b * CN + c] = acc;
  }
}

// ---------------------------------------------------------------------------
// Launch
// Inputs (setup order): 0:x 1:v 2:phase 3:W_re 4:W_im 5:wf_re 6:wf_im 7:lin_w 8:lin_b
// ---------------------------------------------------------------------------
extern "C" void kernel_launch(void* const* d_in, const int* in_sizes, int n_in,
                              void* d_out, int out_size, void* d_ws, size_t ws_size,
                              hipStream_t stream) {
  const float* x      = (const float*)d_in[0];
  const float* v      = (const float*)d_in[1];
  const float* phase  = (const float*)d_in[2];
  const float* W_re   = (const float*)d_in[3];
  const float* W_im   = (const float*)d_in[4];
  const float* wf_re  = (const float*)d_in[5];
  const float* wf_im  = (const float*)d_in[6];
  const float* lin_w  = (const float*)d_in[7];
  const float* lin_b  = (const float*)d_in[8];
  float* out = (float*)d_out;

  // Workspace layout (floats)
  float* ws    = (float*)d_ws;
  float* WF_re = ws;                        // 200
  float* WF_im = WF_re + TN;                // 200
  float* G_re  = WF_im + TN;                // 40000
  float* G_im  = G_re + TN * TN;            // 40000
  float* Kmat  = G_im + TN * TN;            // 784*208 = 163072
  float* realB = Kmat + (size_t)PN * TP;    // 1024*200 = 204800
  // total ~448272 floats (~1.8 MB)

  // 1) WF = FFT(wf)
  wf_fft_kernel<<<1, 256, 0, stream>>>(wf_re, wf_im, WF_re, WF_im);

  // 2) G table
  {
    int n = TN * TN;
    build_G_kernel<<<(n + 255) / 256, 256, 0, stream>>>(WF_re, WF_im, G_re, G_im);
  }

  // 3) K = A_re @ G_re - A_im @ G_im (padded to 208 cols)
  build_K_kernel<<<PN, 256, 0, stream>>>(v, phase, W_re, W_im, G_re, G_im, Kmat);

  // 4) real = X @ K  (WMMA fp32)
  {
    dim3 grid(BN / 16, TP / 16);   // 64 x 13
    gemm_wmma_kernel<<<grid, 32, 0, stream>>>(x, Kmat, realB);
  }

  // 5) features + linear
  feats_linear_kernel<<<(BN + 255) / 256, 256, 0, stream>>>(realB, lin_w, lin_b, out);
}